// weightup_67345087201384
// MI455X (gfx1250) — compile-verified
//
#include <hip/hip_runtime.h>
#include <math.h>

typedef __attribute__((ext_vector_type(2))) float v2f;
typedef __attribute__((ext_vector_type(8))) float v8f;

#define MDIM   1024
#define BSROWS 65536
#define NSTEPS 199
#define ALR    0.1f
#define AB1    0.9f
#define AB2    0.999f
#define AEPS   1e-8f

// ---- persistent-kernel config ----
#define NB2  16            // blocks in adam kernel (co-resident)
#define TPB2 512           // threads per block
#define RPB  64            // rows per block  (MDIM / NB2)
#define TPR  8             // threads per row (TPB2 / RPB)
#define KSEG 128           // K elements per thread (MDIM / TPR)

// ---------------------------------------------------------------------------
// Kernel 1: Q = 0.5*(va + va^T); a0 = 1 - mean; zero barrier words
// ---------------------------------------------------------------------------
__global__ void init_kernel(const float* __restrict__ mean,
                            const float* __restrict__ va,
                            float* __restrict__ Q,
                            float* __restrict__ a,
                            unsigned* __restrict__ bar) {
  const int stride = gridDim.x * blockDim.x;
  for (int idx = blockIdx.x * blockDim.x + threadIdx.x; idx < MDIM * MDIM; idx += stride) {
    const int i = idx >> 10;
    const int j = idx & (MDIM - 1);
    Q[idx] = 0.5f * (va[idx] + va[j * MDIM + i]);
  }
  const int gid = blockIdx.x * blockDim.x + threadIdx.x;
  if (gid < MDIM) a[gid] = 1.0f - mean[gid];
  if (gid == 0) { bar[0] = 0u; bar[1] = 0u; }
}

// ---------------------------------------------------------------------------
// software grid barrier (all NB2 blocks resident)
// ---------------------------------------------------------------------------
__device__ __forceinline__ void grid_barrier(unsigned* cnt, unsigned* gen,
                                             unsigned nblocks, unsigned target) {
  __syncthreads();
  if (threadIdx.x == 0) {
    unsigned prev = __hip_atomic_fetch_add(cnt, 1u, __ATOMIC_ACQ_REL,
                                           __HIP_MEMORY_SCOPE_AGENT);
    if (prev == nblocks - 1u) {
      __hip_atomic_store(cnt, 0u, __ATOMIC_RELAXED, __HIP_MEMORY_SCOPE_AGENT);
      __hip_atomic_store(gen, target, __ATOMIC_RELEASE, __HIP_MEMORY_SCOPE_AGENT);
    } else {
      while (__hip_atomic_load(gen, __ATOMIC_ACQUIRE,
                               __HIP_MEMORY_SCOPE_AGENT) < target) {
        __builtin_amdgcn_s_sleep(2);
      }
    }
  }
  __syncthreads();
}

// ---------------------------------------------------------------------------
// Kernel 2: 199 Adam steps, persistent. Q cached in registers (128 f/thread),
// a-vector exchanged through L2, staged per-step into LDS.
// ---------------------------------------------------------------------------
__global__ void __launch_bounds__(TPB2, 1)
adam_kernel(const float* __restrict__ mean,
            const float* __restrict__ Q,
            float* __restrict__ g_a,
            unsigned* __restrict__ bar) {
  __shared__ float a_s[MDIM];
  __shared__ float psum[TPB2];

  const int tid  = threadIdx.x;
  const int part = tid & (TPR - 1);       // which K segment
  const int irow = tid >> 3;              // local row 0..63
  const int row  = blockIdx.x * RPB + irow;
  const int koff = part * KSEG;

  // one-time load of this thread's Q slice into registers (32 x float4)
  float4 qv[KSEG / 4];
  {
    const float4* qp = reinterpret_cast<const float4*>(Q + row * MDIM + koff);
#pragma unroll
    for (int j = 0; j < KSEG / 4; ++j) qv[j] = qp[j];
  }

  // per-row Adam state lives in the registers of threads tid < RPB
  float w = 1.0f, mo = 0.0f, vo = 0.0f, b1p = 1.0f, b2p = 1.0f;
  float mrow = 0.0f;
  if (tid < RPB) mrow = mean[blockIdx.x * RPB + tid];

  for (int step = 0; step < NSTEPS; ++step) {
    // stage a = w - m into LDS
    for (int j = tid; j < MDIM; j += TPB2) a_s[j] = g_a[j];
    __syncthreads();

    // partial dot of my Q row segment against a
    const float4* as4 = reinterpret_cast<const float4*>(&a_s[koff]);
    float acc = 0.0f;
#pragma unroll
    for (int j = 0; j < KSEG / 4; ++j) {
      float4 q = qv[j], x = as4[j];
      acc += q.x * x.x + q.y * x.y + q.z * x.z + q.w * x.w;
    }
    psum[tid] = acc;
    __syncthreads();

    if (tid < RPB) {
      float g = 0.0f;
#pragma unroll
      for (int k = 0; k < TPR; ++k) g += psum[tid * TPR + k];
      b1p *= AB1;
      b2p *= AB2;
      mo = AB1 * mo + (1.0f - AB1) * g;
      vo = AB2 * vo + (1.0f - AB2) * g * g;
      const float mhat = mo / (1.0f - b1p);
      const float vhat = vo / (1.0f - b2p);
      w -= ALR * mhat / (sqrtf(vhat) + AEPS);
      g_a[blockIdx.x * RPB + tid] = w - mrow;   // publish new a element
    }
    grid_barrier(&bar[0], &bar[1], NB2, (unsigned)(step + 1));
  }
}

// ---------------------------------------------------------------------------
// Kernel 3: p3[blk] = partial of a^T Q a   (4 blocks x 256 thr, 1 row/thread)
// ---------------------------------------------------------------------------
__global__ void quad_kernel(const float* __restrict__ Q,
                            const float* __restrict__ a,
                            float* __restrict__ p3) {
  __shared__ float a_s[MDIM];
  __shared__ float psum[256];
  const int tid = threadIdx.x;
  for (int j = tid; j < MDIM; j += 256) a_s[j] = a[j];
  __syncthreads();

  const int row = blockIdx.x * 256 + tid;
  const float4* qp = reinterpret_cast<const float4*>(Q + row * MDIM);
  const float4* ap = reinterpret_cast<const float4*>(a_s);
  float dot = 0.0f;
#pragma unroll 8
  for (int j = 0; j < MDIM / 4; ++j) {
    float4 q = qp[j], x = ap[j];
    dot += q.x * x.x + q.y * x.y + q.z * x.z + q.w * x.w;
  }
  psum[tid] = a_s[row] * dot;
  __syncthreads();
  for (int off = 128; off > 0; off >>= 1) {
    if (tid < off) psum[tid] += psum[tid + off];
    __syncthreads();
  }
  if (tid == 0) p3[blockIdx.x] = psum[0];
}

// ---------------------------------------------------------------------------
// Kernel 4: p4[blk] = partial of sum(log_sigmoid(xt @ mean)) using
// v_wmma_f32_16x16x4_f32. Each wave handles a 16-row strip of xt.
//
// Column->K-slot mapping trick: the hardware K-slot <-> lane/vgpr mapping is
// identical for A and B, and the dot product is invariant under permuting k,
// so we may assign data columns to K slots arbitrarily as long as A and B
// agree. Each lane does ONE global_load_b128 (float4): lower half-wave lanes
// carry columns kb..kb+3, upper half-wave lanes carry kb+4..kb+7. Two WMMAs
// per load consume (.x,.y) and (.z,.w); B is built from the identically
// permuted mean chunk (only column 0 nonzero, masked branch-free with a 0/1
// multiply -> no EXEC-masked LDS loads). D column 0 accumulates the 16 dot
// products in full fp32 (lane 0 vgprs 0-7 = rows 0-7; lane 16 = rows 8-15).
// ---------------------------------------------------------------------------
__device__ __forceinline__ float log_sigmoid(float x) {
  return fminf(x, 0.0f) - log1pf(expf(-fabsf(x)));
}

__global__ void __launch_bounds__(256, 2)
logsig_kernel(const float* __restrict__ mean,
              const float* __restrict__ xt,
              float* __restrict__ p4) {
  __shared__ float m_s[MDIM];
  __shared__ float psum[256];
  const int tid  = threadIdx.x;
  const int lane = tid & 31;
  const int wv   = tid >> 5;                       // 8 waves per block
  for (int j = tid; j < MDIM; j += 256) m_s[j] = mean[j];
  __syncthreads();

  const int strip = blockIdx.x * 8 + wv;           // 4096 strips total
  const int row   = strip * 16 + (lane & 15);
  const int hi4   = (lane >> 4) << 2;              // 0 or 4 (K half select)
  const float colf = ((lane & 15) == 0) ? 1.0f : 0.0f;  // only col 0 of B

  const float* arow = xt + (size_t)row * MDIM + hi4;
  v8f acc = {};
#pragma unroll 4
  for (int kb = 0; kb < MDIM; kb += 8) {
    __builtin_prefetch((const void*)(arow + kb + 128), 0, 1);
    const float4 av = *reinterpret_cast<const float4*>(arow + kb);
    const float4 mh = *reinterpret_cast<const float4*>(&m_s[kb + hi4]);
    v2f a0, a1, b0, b1;
    a0.x = av.x;        a0.y = av.y;
    a1.x = av.z;        a1.y = av.w;
    b0.x = colf * mh.x; b0.y = colf * mh.y;
    b1.x = colf * mh.z; b1.y = colf * mh.w;
    acc = __builtin_amdgcn_wmma_f32_16x16x4_f32(
        false, a0, false, b0, (short)0, acc, false, false);
    acc = __builtin_amdgcn_wmma_f32_16x16x4_f32(
        false, a1, false, b1, (short)0, acc, false, false);
  }

  // lanes 0 and 16 hold column 0 of D = the 16 dot products of this strip
  float s = 0.0f;
  if ((lane & 15) == 0) {
#pragma unroll
    for (int v = 0; v < 8; ++v) s += log_sigmoid(acc[v]);
  }
  psum[tid] = s;
  __syncthreads();
  for (int off = 128; off > 0; off >>= 1) {
    if (tid < off) psum[tid] += psum[tid + off];
    __syncthreads();
  }
  if (tid == 0) p4[blockIdx.x] = psum[0];
}

// ---------------------------------------------------------------------------
// Kernel 5: deterministic fixed-order finalize.
// out = -(inter1 + inter2) = 0.5*a^T Q a - sum(log_sigmoid)
// ---------------------------------------------------------------------------
__global__ void final_kernel(const float* __restrict__ p3,
                             const float* __restrict__ p4,
                             float* __restrict__ out) {
  if (threadIdx.x == 0 && blockIdx.x == 0) {
    float aqa = 0.0f;
    for (int i = 0; i < 4; ++i) aqa += p3[i];
    float ls = 0.0f;
    for (int i = 0; i < 512; ++i) ls += p4[i];
    out[0] = 0.5f * aqa - ls;
  }
}

// ---------------------------------------------------------------------------
extern "C" void kernel_launch(void* const* d_in, const int* in_sizes, int n_in,
                              void* d_out, int out_size, void* d_ws, size_t ws_size,
                              hipStream_t stream) {
  const float* mean = (const float*)d_in[0];   // [1024]
  const float* va   = (const float*)d_in[1];   // [1024,1024]
  const float* xt   = (const float*)d_in[2];   // [65536,1024]
  float* out = (float*)d_out;

  // workspace layout (floats): Q[M*M] | a[M] | p3[4] | p4[512] | bar[2]
  float*    Q   = (float*)d_ws;
  float*    a   = Q + (size_t)MDIM * MDIM;
  float*    p3  = a + MDIM;
  float*    p4  = p3 + 4;
  unsigned* bar = (unsigned*)(p4 + 512);

  init_kernel<<<1024, 256, 0, stream>>>(mean, va, Q, a, bar);
  adam_kernel<<<NB2, TPB2, 0, stream>>>(mean, Q, a, bar);
  quad_kernel<<<4, 256, 0, stream>>>(Q, a, p3);
  logsig_kernel<<<512, 256, 0, stream>>>(mean, xt, p4);
  final_kernel<<<1, 64, 0, stream>>>(p3, p4, out);
}